// DipoleGrid_16329465659904
// MI455X (gfx1250) — compile-verified
//
#include <hip/hip_runtime.h>

// CDNA5 (gfx1250) dipole-grid torque via fp32 WMMA.
// field_c[i] = sum_j K_c[i][j] * m_c[j],  K_c[i][j] = (1/4pi)*(3*d_c^2 - r^2)/r^5
// Integer-lattice positions => r^2 is a small exact integer, so the radial
// factor s(r^2) = (1/4pi)*r^-5 is precomputed into an LDS table (7939 entries,
// tbl[0]=0 masks the self-pair with no branch). Accumulation runs through
// V_WMMA_F32_16X16X4_F32 (full fp32 matrix path).

typedef __attribute__((ext_vector_type(2))) float v2f;
typedef __attribute__((ext_vector_type(8))) float v8f;

#define NX 64
#define NY 64
#define NSITES (NX * NY)
#define WAVES_PER_BLOCK 8
#define R2MAX (2 * (NX - 1) * (NX - 1)) /* 7938 */
#define TBLN (R2MAX + 1)
#define C4PI 0.07957747154594767f /* 1/(4*pi), MU0 = 1 */

__global__ __launch_bounds__(32 * WAVES_PER_BLOCK)
void dipole_torque_wmma(const float* __restrict__ m,
                        const float* __restrict__ pos,
                        const float* __restrict__ ext,
                        float* __restrict__ out) {
  extern __shared__ float lds[];
  float* sm  = lds;                  // staged moments   [NSITES*2]  32 KB
  float* sp  = sm + NSITES * 2;      // staged positions [NSITES*2]  32 KB
  float* tbl = sp + NSITES * 2;      // radial table     [TBLN]      31.7 KB
  float* sf  = tbl + TBLN;           // per-wave field scratch [WAVES*16*2]

  const int tid = threadIdx.x;

  // Stage inputs and build the r^-5 table (precise sqrt/div done 8K times
  // per block (254K in all) instead of 16.8M times in the pair loop).
  for (int k = tid; k < NSITES * 2; k += blockDim.x) {
    sm[k] = m[k];
    sp[k] = pos[k];
  }
  for (int k = tid; k < TBLN; k += blockDim.x) {
    const float r2f = (float)k;
    tbl[k] = (k == 0) ? 0.0f : (C4PI / (r2f * r2f * sqrtf(r2f)));
  }
  __syncthreads();

  const int lane = tid & 31;
  const int wave = tid >> 5;
  const int tile = blockIdx.x * WAVES_PER_BLOCK + wave; // 0..255
  const int i0   = tile * 16;
  const int mrow = lane & 15;        // M row this lane supplies in A
  const int i    = i0 + mrow;        // destination site index
  const int tsel = lane >> 4;        // 0 -> K slots {0,1}; 1 -> K slots {2,3}

  const float xi = sp[2 * i + 0];
  const float yi = sp[2 * i + 1];

  v8f acc = {}; // 16x16 f32 accumulator; col 0 = field_x, col 1 = field_y

  // K=4 per WMMA packs two source sites: (j0 -> slots 0,1), (j0+1 -> slots 2,3).
#pragma unroll 4
  for (int j0 = 0; j0 < NSITES; j0 += 2) {
    const int jt = j0 + tsel;
    const float xj = sp[2 * jt + 0];
    const float yj = sp[2 * jt + 1];
    const float dx = xi - xj;
    const float dy = yi - yj;
    const float r2 = fmaf(dx, dx, dy * dy);      // exact small integer
    const int  idx = (int)(r2 + 0.5f);
    const float s  = tbl[idx];                   // (1/4pi)*r^-5 ; 0 for self-pair

    // A 16x4 layout: lanes 0-15 hold M=lane, VGPR0=K0, VGPR1=K1;
    //                lanes 16-31 hold M=lane-16, VGPR0=K2, VGPR1=K3.
    v2f a;
    a.x = s * fmaf(3.0f * dx, dx, -r2); // K_x(i, jt)
    a.y = s * fmaf(3.0f * dy, dy, -r2); // K_y(i, jt)

    // B 4x16 layout: VGPR0 = rows K0 (lanes 0-15) / K2 (lanes 16-31),
    //                VGPR1 = rows K1 / K3; N = lane & 15.
    const float mxj = sm[2 * jt + 0];
    const float myj = sm[2 * jt + 1];
    v2f b;
    b.x = (mrow == 0) ? mxj : 0.0f; // column 0 <- m_x[jt]
    b.y = (mrow == 1) ? myj : 0.0f; // column 1 <- m_y[jt]

    acc = __builtin_amdgcn_wmma_f32_16x16x4_f32(
        /*neg_a=*/false, a, /*neg_b=*/false, b,
        /*c_mod=*/(short)0, acc, /*reuse_a=*/false, /*reuse_b=*/false);
  }

  // C/D layout: VGPR v -> lanes 0-15: (M=v, N=lane); lanes 16-31: (M=v+8, N=lane-16).
  float* myf = sf + wave * 32; // [16][2]
  if (mrow == 0) {             // lanes 0 and 16 hold column 0 (field_x)
#pragma unroll
    for (int v = 0; v < 8; ++v) myf[2 * (tsel * 8 + v) + 0] = acc[v];
  }
  if (mrow == 1) {             // lanes 1 and 17 hold column 1 (field_y)
#pragma unroll
    for (int v = 0; v < 8; ++v) myf[2 * (tsel * 8 + v) + 1] = acc[v];
  }
  __syncthreads();

  if (lane < 16) {
    const float fx = myf[2 * lane + 0] + ext[2 * i + 0];
    const float fy = myf[2 * lane + 1] + ext[2 * i + 1];
    out[i] = sm[2 * i + 0] * fy - sm[2 * i + 1] * fx; // 2D cross, z-component
  }
}

extern "C" void kernel_launch(void* const* d_in, const int* in_sizes, int n_in,
                              void* d_out, int out_size, void* d_ws, size_t ws_size,
                              hipStream_t stream) {
  const float* m   = (const float*)d_in[0];
  const float* pos = (const float*)d_in[1];
  const float* ext = (const float*)d_in[2];
  float* out = (float*)d_out;

  dim3 grid(NSITES / (16 * WAVES_PER_BLOCK)); // 32 blocks
  dim3 block(32 * WAVES_PER_BLOCK);           // 8 waves each, wave32
  size_t shmem =
      (size_t)(NSITES * 4 + TBLN + WAVES_PER_BLOCK * 32) * sizeof(float);

  hipLaunchKernelGGL(dipole_torque_wmma, grid, block, shmem, stream,
                     m, pos, ext, out);
}